// GCN_Graph_34497177322038
// MI455X (gfx1250) — compile-verified
//
#include <hip/hip_runtime.h>

typedef float v2f __attribute__((ext_vector_type(2)));
typedef float v8f __attribute__((ext_vector_type(8)));

#define NN 100000
#define NE 1600000
#define NG 256
#define FDH 64
#define FDE 128

// ---------------- degree / norm ----------------
__global__ void k_init_deg(float* __restrict__ deg, int n) {
  int i = blockIdx.x * blockDim.x + threadIdx.x;
  if (i < n) deg[i] = 1.0f;                       // self-loop contributes 1
}

__global__ void k_count_deg(const long long* __restrict__ dst,
                            float* __restrict__ deg, int e) {
  int i = blockIdx.x * blockDim.x + threadIdx.x;
  if (i < e) atomicAdd(&deg[(int)dst[i]], 1.0f);
}

__global__ void k_dinv(float* __restrict__ deg, int n) {
  int i = blockIdx.x * blockDim.x + threadIdx.x;
  if (i < n) deg[i] = rsqrtf(deg[i]);             // deg >= 1 always
}

// ---------------- fp32 WMMA GEMM: out[M,NC] = A[M,K] @ W[NC,K]^T (+bias) ----------------
// One wave computes one 16x16 output tile via V_WMMA_F32_16X16X4_F32.
__global__ void k_gemm_wmma(const float* __restrict__ A, const float* __restrict__ W,
                            const float* __restrict__ bias, float* __restrict__ out,
                            int M, int K, int NC) {
  int gtid   = blockIdx.x * blockDim.x + threadIdx.x;
  int wave   = gtid >> 5;
  int lane   = gtid & 31;
  int tilesx = NC >> 4;
  int tiles  = (M >> 4) * tilesx;
  if (wave >= tiles) return;                      // whole-wave exit: EXEC stays all-1s
  int row0 = (wave / tilesx) << 4;
  int col0 = (wave % tilesx) << 4;
  int m    = lane & 15;
  int hi   = lane >> 4;                           // 0: K,K+1   1: K+2,K+3

  const float* arow = A + (size_t)(row0 + m) * K; // A fragment source (row m)
  const float* wrow = W + (size_t)(col0 + m) * K; // B[k][n] = W[n][k], n = col0+m

  v8f c = {};
  for (int k0 = 0; k0 < K; k0 += 4) {
    v2f a, b;
    a.x = arow[k0 + 2 * hi];
    a.y = arow[k0 + 2 * hi + 1];
    b.x = wrow[k0 + 2 * hi];
    b.y = wrow[k0 + 2 * hi + 1];
    c = __builtin_amdgcn_wmma_f32_16x16x4_f32(false, a, false, b, (short)0, c,
                                              false, false);
  }

  int ncol  = col0 + m;
  int mbase = row0 + (hi ? 8 : 0);                // D: VGPR r -> row r (lanes 0-15) / r+8
  float badd = bias ? bias[ncol] : 0.0f;
#pragma unroll
  for (int r = 0; r < 8; ++r)
    out[(size_t)(mbase + r) * NC + ncol] = c[r] + badd;
}

// ---------------- aggregation ----------------
__global__ void k_agg_init(const float* __restrict__ hW, const float* __restrict__ dinv,
                           const float* __restrict__ bvec, float* __restrict__ agg, int n) {
  int gid = blockIdx.x * blockDim.x + threadIdx.x;
  if (gid >= n * FDH) return;
  int node = gid >> 6, f = gid & 63;
  float d = dinv[node];
  agg[gid] = hW[gid] * d * d + bvec[f];           // self-loop term + bias
}

__global__ void k_edge_scatter(const long long* __restrict__ src,
                               const long long* __restrict__ dst,
                               const float* __restrict__ dinv,
                               const float* __restrict__ hW,
                               float* __restrict__ agg, int e) {
  int gid = blockIdx.x * blockDim.x + threadIdx.x;
  if (gid >= e * FDH) return;                     // e*64 = 102.4M < 2^31
  int ei = gid >> 6, f = gid & 63;
  int s = (int)src[ei], d = (int)dst[ei];
  float w = dinv[s] * dinv[d];
  atomicAdd(&agg[(size_t)d * FDH + f], hW[(size_t)s * FDH + f] * w);
}

// ---------------- layernorm + leaky relu (in place) ----------------
__global__ void k_ln_leaky(float* __restrict__ h, const float* __restrict__ gamma,
                           const float* __restrict__ beta, int n) {
  int node = blockIdx.x * blockDim.x + threadIdx.x;
  if (node >= n) return;
  float* row = h + (size_t)node * FDH;
  float4 buf[16];
  const float4* r4 = (const float4*)row;
  float s = 0.0f;
#pragma unroll
  for (int i = 0; i < 16; ++i) {
    buf[i] = r4[i];
    s += buf[i].x + buf[i].y + buf[i].z + buf[i].w;
  }
  float mu = s * (1.0f / 64.0f);
  float s2 = 0.0f;
#pragma unroll
  for (int i = 0; i < 16; ++i) {
    float a = buf[i].x - mu, b = buf[i].y - mu, cc = buf[i].z - mu, d = buf[i].w - mu;
    s2 += a * a + b * b + cc * cc + d * d;
  }
  float rstd = rsqrtf(s2 * (1.0f / 64.0f) + 1e-5f);
  float4* w4 = (float4*)row;
#pragma unroll
  for (int i = 0; i < 16; ++i) {
    float o[4] = {buf[i].x, buf[i].y, buf[i].z, buf[i].w};
#pragma unroll
    for (int j = 0; j < 4; ++j) {
      int f = i * 4 + j;
      float t = (o[j] - mu) * rstd * gamma[f] + beta[f];
      o[j] = t > 0.0f ? t : 0.1f * t;
    }
    w4[i] = make_float4(o[0], o[1], o[2], o[3]);
  }
}

// ---------------- pooling ----------------
__device__ __forceinline__ unsigned fenc(float f) {
  unsigned u = __float_as_uint(f);
  return (u & 0x80000000u) ? ~u : (u | 0x80000000u);
}
__device__ __forceinline__ float fdec(unsigned u) {
  return (u & 0x80000000u) ? __uint_as_float(u ^ 0x80000000u) : __uint_as_float(~u);
}

__global__ void k_pool_init(unsigned* __restrict__ gmp, float* __restrict__ sums,
                            float* __restrict__ counts) {
  int gid = blockIdx.x * blockDim.x + threadIdx.x;
  if (gid < NG * FDH) {
    gmp[gid]  = fenc(-__builtin_inff());          // segment_max identity
    sums[gid] = 0.0f;
  }
  if (gid < NG) counts[gid] = 0.0f;
}

__global__ void k_pool_scatter(const float* __restrict__ h,
                               const long long* __restrict__ batch,
                               unsigned* __restrict__ gmp, float* __restrict__ sums,
                               float* __restrict__ counts, int n) {
  int gid = blockIdx.x * blockDim.x + threadIdx.x;
  if (gid >= n * FDH) return;
  int node = gid >> 6, f = gid & 63;
  int g = (int)batch[node];
  float v = h[gid];
  atomicMax(&gmp[g * FDH + f], fenc(v));
  atomicAdd(&sums[g * FDH + f], v);
  if (f == 0) atomicAdd(&counts[g], 1.0f);
}

__global__ void k_pool_fin(const unsigned* __restrict__ gmp, const float* __restrict__ sums,
                           const float* __restrict__ counts, float* __restrict__ pooled) {
  int gid = blockIdx.x * blockDim.x + threadIdx.x;
  if (gid >= NG * FDH) return;
  int g = gid >> 6, f = gid & 63;
  pooled[g * (2 * FDH) + f]       = fdec(gmp[gid]);
  pooled[g * (2 * FDH) + FDH + f] = sums[gid] / fmaxf(counts[g], 1.0f);
}

// ---------------- driver ----------------
extern "C" void kernel_launch(void* const* d_in, const int* in_sizes, int n_in,
                              void* d_out, int out_size, void* d_ws, size_t ws_size,
                              hipStream_t stream) {
  const float*     x     = (const float*)d_in[0];
  const long long* eidx  = (const long long*)d_in[1];   // [2, E] int64
  const long long* batch = (const long long*)d_in[2];   // [N]    int64
  const float* Wl[3] = {(const float*)d_in[3], (const float*)d_in[7],  (const float*)d_in[11]};
  const float* bl[3] = {(const float*)d_in[4], (const float*)d_in[8],  (const float*)d_in[12]};
  const float* gl[3] = {(const float*)d_in[5], (const float*)d_in[9],  (const float*)d_in[13]};
  const float* tl[3] = {(const float*)d_in[6], (const float*)d_in[10], (const float*)d_in[14]};
  const float* Wlin = (const float*)d_in[15];
  const float* blin = (const float*)d_in[16];
  float* out = (float*)d_out;

  char* ws = (char*)d_ws;
  float*    dinv   = (float*)ws;                  ws += (size_t)NN * 4;
  float*    bufA   = (float*)ws;                  ws += (size_t)NN * FDH * 4;   // agg / h
  float*    bufB   = (float*)ws;                  ws += (size_t)NN * FDH * 4;   // h @ W^T
  unsigned* gmp    = (unsigned*)ws;               ws += (size_t)NG * FDH * 4;
  float*    sums   = (float*)ws;                  ws += (size_t)NG * FDH * 4;
  float*    counts = (float*)ws;                  ws += (size_t)NG * 4;
  float*    pooled = (float*)ws;                  /* NG * 2*FDH floats */

  const long long* src = eidx;
  const long long* dst = eidx + NE;

  const int B = 256;
  // degree -> dinv
  k_init_deg <<<(NN + B - 1) / B, B, 0, stream>>>(dinv, NN);
  k_count_deg<<<(NE + B - 1) / B, B, 0, stream>>>(dst, dinv, NE);
  k_dinv     <<<(NN + B - 1) / B, B, 0, stream>>>(dinv, NN);

  const float* hin = x;
  int gtiles = (NN / 16) * (FDH / 16);            // 25000 tiles, 8 waves/block
  for (int l = 0; l < 3; ++l) {
    k_gemm_wmma   <<<(gtiles + 7) / 8, B, 0, stream>>>(hin, Wl[l], nullptr, bufB,
                                                       NN, FDH, FDH);
    k_agg_init    <<<((size_t)NN * FDH + B - 1) / B, B, 0, stream>>>(bufB, dinv, bl[l],
                                                                     bufA, NN);
    k_edge_scatter<<<((size_t)NE * FDH + B - 1) / B, B, 0, stream>>>(src, dst, dinv,
                                                                     bufB, bufA, NE);
    k_ln_leaky    <<<(NN + B - 1) / B, B, 0, stream>>>(bufA, gl[l], tl[l], NN);
    hin = bufA;
  }

  k_pool_init   <<<(NG * FDH + B - 1) / B, B, 0, stream>>>(gmp, sums, counts);
  k_pool_scatter<<<((size_t)NN * FDH + B - 1) / B, B, 0, stream>>>(bufA, batch, gmp,
                                                                   sums, counts, NN);
  k_pool_fin    <<<(NG * FDH + B - 1) / B, B, 0, stream>>>(gmp, sums, counts, pooled);

  // final linear: [256,128] @ Wlin[128,128]^T + blin, 128 tiles -> 16 blocks
  int ftiles = (NG / 16) * (FDE / 16);
  k_gemm_wmma<<<(ftiles + 7) / 8, B, 0, stream>>>(pooled, Wlin, blin, out,
                                                  NG, 2 * FDH, FDE);
}